// GNNLayer_31207232372898
// MI455X (gfx1250) — compile-verified
//
#include <hip/hip_runtime.h>

typedef float v2f __attribute__((ext_vector_type(2)));
typedef float v8f __attribute__((ext_vector_type(8)));

#define N_B   2
#define V_N   10000
#define E_N   320000
#define ND    128
#define OD    128
#define IN_DIM 384

// LDS layout (floats).
// Weights are stored PAIR-INTERLEAVED: pair-row k2 holds {W[2k2][c], W[2k2+1][c]}
// contiguously per column -> one ds_load_b64 per B fragment, no repack movs.
// Pair-row stride PS=272 (=256+16) offsets the two half-waves' rows by 16 banks.
#define PS 272
#define HS 132                                   // h tile row stride (bank-clean)
#define SW1_OFF 0                                // 192 pair-rows * 272 = 52224
#define SW2_OFF (192 * PS)
#define SH_OFF  (SW2_OFF + 64 * PS)              // + 64 * 272 = 17408
#define SMEM_FLOATS (SH_OFF + 4 * 16 * HS)       // + 8448 = 78080 floats = 312320 B

// ---------------------------------------------------------------------------
// degree[j] = max(1, sum_i A[i, j])
// ---------------------------------------------------------------------------
__global__ void degree_kernel(const float* __restrict__ A, float* __restrict__ deg) {
    int j = blockIdx.x * blockDim.x + threadIdx.x;
    if (j >= V_N) return;
    float s = 0.0f;
    for (int i = 0; i < V_N; ++i) s += A[(size_t)i * V_N + j];
    deg[j] = s < 1.0f ? 1.0f : s;
}

// ---------------------------------------------------------------------------
// Fused edge MLP: x = [v_src | v_dst | e] (384) -> relu(x*W1+b1) -> *W2+b2
// MODE 0: scatter-add y into accum[n, dst, :]   (node update path)
// MODE 1: out_e = relu(y + e)                   (edge update path)
// Block = 128 threads = 4 waves; each wave owns a 16-row x 128-col tile.
// ---------------------------------------------------------------------------
template <int MODE>
__global__ __launch_bounds__(128, 1)
void edge_mlp_kernel(const float* __restrict__ v,
                     const float* __restrict__ e,
                     const float* __restrict__ W1,
                     const float* __restrict__ B1,
                     const float* __restrict__ W2,
                     const float* __restrict__ B2,
                     const int*   __restrict__ G,
                     float* __restrict__ accum,
                     float* __restrict__ outE) {
    extern __shared__ float smem[];
    float* sW1 = smem + SW1_OFF;    // pair-interleaved, 192 pair-rows
    float* sW2 = smem + SW2_OFF;    // pair-interleaved,  64 pair-rows
    float* sH  = smem + SH_OFF;

    const int tid = threadIdx.x;

    // ---- stage weights into LDS, pair-interleaving rows k and k+1 ----
    for (int i = tid; i < 192 * 128; i += blockDim.x) {     // W1: 24576 pairs
        int k2 = i >> 7, c = i & 127;
        v2f p;
        p.x = W1[(2 * k2)     * OD + c];
        p.y = W1[(2 * k2 + 1) * OD + c];
        *(v2f*)(sW1 + k2 * PS + 2 * c) = p;
    }
    for (int i = tid; i < 64 * 128; i += blockDim.x) {      // W2: 8192 pairs
        int k2 = i >> 7, c = i & 127;
        v2f p;
        p.x = W2[(2 * k2)     * OD + c];
        p.y = W2[(2 * k2 + 1) * OD + c];
        *(v2f*)(sW2 + k2 * PS + 2 * c) = p;
    }
    __syncthreads();

    const int wave = tid >> 5;
    const int lane = tid & 31;
    const int m    = lane & 15;          // tile row (A) / tile col (B,C)
    const int hi   = lane >> 4;          // half-wave: owns k-pair 0/1 or 2/3
    const int kk   = hi << 1;
    float* myH = sH + wave * (16 * HS);

    // per-lane biases (column j*16+m), tile-invariant -> registers
    float b1r[8], b2r[8];
#pragma unroll
    for (int j = 0; j < 8; ++j) {
        b1r[j] = B1[j * 16 + m];
        b2r[j] = B2[j * 16 + m];
    }

    const int numTiles = (N_B * E_N) / 16;   // 40000, batch boundary tile-aligned
    const v8f vzero = {0, 0, 0, 0, 0, 0, 0, 0};

    for (int t = blockIdx.x * 4 + wave; t < numTiles; t += gridDim.x * 4) {
        const int r0 = t * 16;
        const int n  = r0 / E_N;
        const int e0 = r0 - n * E_N;

        const int myEdge = e0 + m;
        const int src = G[2 * myEdge + 0];
        const int dst = G[2 * myEdge + 1];

        // per-lane row base pointers for the three 128-wide feature segments
        const float* pa = v + ((size_t)n * V_N + src) * ND;      // v_src
        const float* pb = v + ((size_t)n * V_N + dst) * ND;      // v_dst
        const float* pc = e + ((size_t)n * E_N + myEdge) * ND;   // e

        v8f acc[8];
#pragma unroll
        for (int j = 0; j < 8; ++j) acc[j] = vzero;

        // ---------- layer 1: K = 384 = 3 phases x 128 ----------
        const float* phases[3] = {pa, pb, pc};
#pragma unroll
        for (int ph = 0; ph < 3; ++ph) {
            const float* p = phases[ph];
            const int pr0 = ph * 64;                       // pair-row base in sW1
            for (int ks = 0; ks < 128; ks += 4) {
                v2f a = *(const v2f*)(p + ks + kk);        // A[m][k..k+1]
                const float* w = sW1 + (size_t)(pr0 + (ks >> 1) + hi) * PS + 2 * m;
#pragma unroll
                for (int j = 0; j < 8; ++j) {
                    v2f b = *(const v2f*)(w + 32 * j);     // {W[k][c], W[k+1][c]}
                    acc[j] = __builtin_amdgcn_wmma_f32_16x16x4_f32(
                        false, a, false, b, (short)0, acc[j], false, false);
                }
            }
        }

        // ---------- bias + relu, park h tile in LDS (16 x 128, stride HS) ----
#pragma unroll
        for (int j = 0; j < 8; ++j) {
            const int c = j * 16 + m;
#pragma unroll
            for (int r = 0; r < 8; ++r) {
                const int row = r + hi * 8;
                float hv = acc[j][r] + b1r[j];
                myH[row * HS + c] = hv > 0.0f ? hv : 0.0f;
            }
        }

        // ---------- layer 2: K = 128, A from LDS h ----------
#pragma unroll
        for (int j = 0; j < 8; ++j) acc[j] = vzero;
        for (int ks = 0; ks < 128; ks += 4) {
            v2f a = *(const v2f*)(myH + m * HS + ks + kk);
            const float* w = sW2 + (size_t)((ks >> 1) + hi) * PS + 2 * m;
#pragma unroll
            for (int j = 0; j < 8; ++j) {
                v2f b = *(const v2f*)(w + 32 * j);
                acc[j] = __builtin_amdgcn_wmma_f32_16x16x4_f32(
                    false, a, false, b, (short)0, acc[j], false, false);
            }
        }

        // ---------- epilogue ----------
        if (MODE == 0) {
#pragma unroll
            for (int j = 0; j < 8; ++j) {
                const int c = j * 16 + m;
#pragma unroll
                for (int r = 0; r < 8; ++r) {
                    const int row = r + hi * 8;
                    const int d = __shfl(dst, row, 32);    // lane 'row' holds dst[row]
                    float y = acc[j][r] + b2r[j];
                    __hip_atomic_fetch_add(&accum[((size_t)n * V_N + d) * OD + c], y,
                                           __ATOMIC_RELAXED, __HIP_MEMORY_SCOPE_AGENT);
                }
            }
        } else {
#pragma unroll
            for (int j = 0; j < 8; ++j) {
                const int c = j * 16 + m;
#pragma unroll
                for (int r = 0; r < 8; ++r) {
                    const int row = r + hi * 8;
                    const size_t off = ((size_t)n * E_N + (e0 + row)) * OD + c;
                    float y = acc[j][r] + b2r[j] + e[off];
                    outE[off] = y > 0.0f ? y : 0.0f;
                }
            }
        }
    }
}

// ---------------------------------------------------------------------------
// out_v = relu(accum / deg + v)
// ---------------------------------------------------------------------------
__global__ void finalize_v_kernel(const float* __restrict__ accum,
                                  const float* __restrict__ v,
                                  const float* __restrict__ deg,
                                  float* __restrict__ outV) {
    size_t idx = (size_t)blockIdx.x * blockDim.x + threadIdx.x;
    const size_t total = (size_t)N_B * V_N * OD;
    if (idx >= total) return;
    int jv = (int)((idx / OD) % V_N);
    float r = accum[idx] / deg[jv] + v[idx];
    outV[idx] = r > 0.0f ? r : 0.0f;
}

// ---------------------------------------------------------------------------
extern "C" void kernel_launch(void* const* d_in, const int* in_sizes, int n_in,
                              void* d_out, int out_size, void* d_ws, size_t ws_size,
                              hipStream_t stream) {
    const float* v   = (const float*)d_in[0];
    const float* e   = (const float*)d_in[1];
    const float* A   = (const float*)d_in[2];
    const float* wn1 = (const float*)d_in[3];
    const float* bn1 = (const float*)d_in[4];
    const float* wn2 = (const float*)d_in[5];
    const float* bn2 = (const float*)d_in[6];
    const float* we1 = (const float*)d_in[7];
    const float* be1 = (const float*)d_in[8];
    const float* we2 = (const float*)d_in[9];
    const float* be2 = (const float*)d_in[10];
    const int*   G   = (const int*)d_in[11];

    float* accum = (float*)d_ws;                         // N*V*OD floats (10.24 MB)
    float* deg   = accum + (size_t)N_B * V_N * OD;       // V floats
    float* outV  = (float*)d_out;                        // N*V*OD
    float* outE  = outV + (size_t)N_B * V_N * OD;        // N*E*OD

    hipMemsetAsync(accum, 0, (size_t)N_B * V_N * OD * sizeof(float), stream);
    degree_kernel<<<(V_N + 255) / 256, 256, 0, stream>>>(A, deg);

    const int smemBytes = SMEM_FLOATS * sizeof(float);
    edge_mlp_kernel<0><<<2000, 128, smemBytes, stream>>>(v, e, wn1, bn1, wn2, bn2, G,
                                                         accum, nullptr);
    edge_mlp_kernel<1><<<2000, 128, smemBytes, stream>>>(v, e, we1, be1, we2, be2, G,
                                                         nullptr, outE);

    const size_t totalV = (size_t)N_B * V_N * OD;
    finalize_v_kernel<<<(int)((totalV + 255) / 256), 256, 0, stream>>>(accum, v, deg, outV);
}